// PrototypeLoss_2035814498849
// MI455X (gfx1250) — compile-verified
//
#include <hip/hip_runtime.h>

// PrototypeLoss on MI455X (gfx1250, wave32).
// loss = mean_n sum_d (features[n,d] - prototypes[labels[n],d])^2
//
// Memory-bound: ~258 MB HBM traffic -> ~11us floor at 23.3 TB/s (prototypes
// are L2-resident: 2 MB vs 192 MB L2). Each wave owns a 16-sample tile and
// uses V_WMMA_F32_16X16X4_F32 with B = all-ones as an exact-f32 row-reduction
// engine: D = A*1 + C accumulates per-row sums of squared diffs in the matrix
// pipe. Since B is all-ones, the result is invariant to K-slot placement, so
// each lane streams its row with 16-byte b128 loads (lanes 0-15 take columns
// [kb,kb+4), lanes 16-31 take [kb+4,kb+8)) and feeds two A-fragments per load.

typedef __attribute__((ext_vector_type(2))) float v2f;
typedef __attribute__((ext_vector_type(4))) float v4f;
typedef __attribute__((ext_vector_type(8))) float v8f;

__global__ void zero_out_kernel(float* out) {
    if (threadIdx.x == 0 && blockIdx.x == 0) out[0] = 0.0f;
}

__global__ __launch_bounds__(256)
void proto_loss_wmma_kernel(const float* __restrict__ features,
                            const int*   __restrict__ labels,
                            const float* __restrict__ prototypes,
                            float*       __restrict__ out,
                            int n_tiles, int D, float inv_n)
{
    const int lane          = threadIdx.x & 31;
    const int wave_in_block = threadIdx.x >> 5;
    const int wave_id       = blockIdx.x * (blockDim.x >> 5) + wave_in_block;
    const int n_waves       = gridDim.x * (blockDim.x >> 5);

    // A-fragment row ownership (CDNA5 ISA): lane l contributes to
    // row M = l & 15. K-slot placement is irrelevant because B == ones.
    const int row  = lane & 15;
    const int koff = (lane >> 4) << 2;   // lanes 0-15: cols+0..3, 16-31: +4..7

    v2f ones; ones.x = 1.0f; ones.y = 1.0f;   // B = 4x16 all-ones

    for (int tile = wave_id; tile < n_tiles; tile += n_waves) {
        const int sample = tile * 16 + row;
        const float* frow = features + (size_t)sample * (size_t)D + koff;
        const int lbl = labels[sample];
        const float* prow = prototypes + (size_t)lbl * (size_t)D + koff;

        v8f c = {0.f, 0.f, 0.f, 0.f, 0.f, 0.f, 0.f, 0.f};

        // 8 columns per row per iteration: one b128 feature load + one b128
        // prototype load -> two WMMA K-steps. D=512 -> 64 iterations.
        for (int kb = 0; kb < D; kb += 8) {
            v4f f = *(const v4f*)(frow + kb);
            v4f p = *(const v4f*)(prow + kb);
            v4f d = f - p;
            v4f q = d * d;

            v2f a0; a0.x = q.x; a0.y = q.y;
            v2f a1; a1.x = q.z; a1.y = q.w;

            // 8 args: (neg_a, A, neg_b, B, c_mod, C, reuse_a, reuse_b)
            c = __builtin_amdgcn_wmma_f32_16x16x4_f32(
                    false, a0, false, ones, (short)0, c, false, false);
            c = __builtin_amdgcn_wmma_f32_16x16x4_f32(
                    false, a1, false, ones, (short)0, c, false, false);
        }

        // Every column of C/D holds the row sum: lane 0 has rows 0-7 in
        // c[0..7], lane 16 has rows 8-15 in c[0..7].
        float s = c[0] + c[1] + c[2] + c[3] + c[4] + c[5] + c[6] + c[7];
        float tile_sum = __shfl(s, 0, 32) + __shfl(s, 16, 32);
        if (lane == 0) {
            atomicAdd(out, tile_sum * inv_n);
        }
    }
}

extern "C" void kernel_launch(void* const* d_in, const int* in_sizes, int n_in,
                              void* d_out, int out_size, void* d_ws, size_t ws_size,
                              hipStream_t stream) {
    const float* features   = (const float*)d_in[0];
    const int*   labels     = (const int*)d_in[1];   // int32 per harness convention
    const float* prototypes = (const float*)d_in[2];
    float*       out        = (float*)d_out;

    const int N = in_sizes[1];          // number of samples (labels count)
    const int D = in_sizes[0] / N;      // feature dim (512)
    const int n_tiles = N / 16;         // N multiple of 16 here (8192 tiles)

    // d_out is poisoned by the harness; zero the accumulator first (in-stream,
    // graph-capture safe).
    zero_out_kernel<<<1, 32, 0, stream>>>(out);

    const int waves_per_block = 8;      // 256 threads = 8 wave32s
    int blocks = (n_tiles + waves_per_block - 1) / waves_per_block;
    if (blocks > 2048) blocks = 2048;
    if (blocks < 1) blocks = 1;

    proto_loss_wmma_kernel<<<blocks, 256, 0, stream>>>(
        features, labels, prototypes, out, n_tiles, D, 1.0f / (float)N);
}